// PoseLoss_49452253446870
// MI455X (gfx1250) — compile-verified
//
#include <hip/hip_runtime.h>

typedef __attribute__((ext_vector_type(2))) float v2f;
typedef __attribute__((ext_vector_type(4))) float v4f;
typedef __attribute__((ext_vector_type(8))) float v8f;

#define DIMS       212
#define SPLIT_COL  66
#define MAX_W1     10.0f
#define MIN_W1     4.0f
#define MAX_E1     2.0f
#define MIN_E1     0.5f
#define MAX_W2     8.0f
#define MIN_W2     3.0f
#define MAX_E2     2.0f
#define MIN_E2     0.5f

#define ROWS_PER_WAVE 16
#define WAVES_PER_WG  8
#define ROWS_PER_WG   (ROWS_PER_WAVE * WAVES_PER_WG)   // 128

// One wave (wave32) handles 16 rows.
// Pass 1: pc[16] via chained V_WMMA_F32_16X16X4_F32 (mat-vec, K chunks of 4).
// Pass 2: wing loss over the 16x212 tile with float4 streaming loads.
__global__ __launch_bounds__(256) void poseloss_main(
    const float* __restrict__ pred,
    const float* __restrict__ targ,
    const float* __restrict__ u,
    const float* __restrict__ mean,
    float* __restrict__ partials)
{
    const int lane = threadIdx.x & 31;
    const int wave = threadIdx.x >> 5;
    const int rowbase = blockIdx.x * ROWS_PER_WG + wave * ROWS_PER_WAVE;

    // ---------------- Pass 1: pc via WMMA f32 16x16x4 ----------------
    // A layout (16x4 f32): lane L<16 -> row L, K = {0,1}; lane L+16 -> row L, K = {2,3}
    // B (4x16): broadcast u chunk into every column -> every column of D = row dot.
    const int half = lane >> 4;          // 0 or 1
    const int mrow = lane & 15;          // my A-matrix row
    const float* trow = targ + (size_t)(rowbase + mrow) * DIMS;

    v8f c = {};
    #pragma unroll 4
    for (int k0 = 0; k0 < DIMS; k0 += 4) {
        const int k = k0 + 2 * half;
        v2f t2 = *(const v2f*)(trow + k);
        v2f m2 = *(const v2f*)(mean + k);
        v2f u2 = *(const v2f*)(u + k);
        v2f a  = t2 - m2;
        // D = A x B + C  (accumulate row-dot chunks)
        c = __builtin_amdgcn_wmma_f32_16x16x4_f32(
                false, a, false, u2, (short)0, c, false, false);
    }
    // c[j] holds dot(row j)   in lanes 0-15,
    //            dot(row j+8) in lanes 16-31.

    // ---------------- Pass 2: wing loss ----------------
    float acc1 = 0.0f;   // sum over cols <  66
    float acc2 = 0.0f;   // sum over cols >= 66
    #pragma unroll
    for (int r = 0; r < ROWS_PER_WAVE; ++r) {
        const float dotv = __shfl(c[r & 7], (r < 8) ? 0 : 16, 32);
        const float pc = fabsf(dotv);

        const float w1 = (MIN_W1 - MAX_W1) * pc + MAX_W1;
        const float e1 = (MIN_E1 - MAX_E1) * pc + MAX_E1;
        const float w2 = (MIN_W2 - MAX_W2) * pc + MAX_W2;
        const float e2 = (MIN_E2 - MAX_E2) * pc + MAX_E2;
        const float ie1 = 1.0f / e1;
        const float ie2 = 1.0f / e2;
        const float C1 = w1 + w1 * __logf(1.0f + w1 * ie1);
        const float C2 = w2 + w2 * __logf(1.0f + w2 * ie2);

        const float* prow = pred + (size_t)(rowbase + r) * DIMS;
        const float* qrow = targ + (size_t)(rowbase + r) * DIMS;

        #pragma unroll
        for (int it = 0; it < 2; ++it) {
            const int cbase = it * 128 + 4 * lane;   // float4 per lane
            if (cbase < DIMS) {
                v4f p = *(const v4f*)(prow + cbase);
                v4f q = *(const v4f*)(qrow + cbase);
                #pragma unroll
                for (int e = 0; e < 4; ++e) {
                    const int col = cbase + e;
                    const float t = fabsf(p[e] - q[e]);
                    const bool first = col < SPLIT_COL;
                    const float w  = first ? w1  : w2;
                    const float ie = first ? ie1 : ie2;
                    const float Cc = first ? C1  : C2;
                    const float wing = (t < w) ? w * __logf(1.0f + t * ie)
                                               : (t - Cc);
                    if (first) acc1 += wing; else acc2 += wing;
                }
            }
        }
    }

    float acc = acc1 * (1.0f / SPLIT_COL) + acc2 * (1.0f / (DIMS - SPLIT_COL));

    // wave reduce
    #pragma unroll
    for (int off = 16; off > 0; off >>= 1)
        acc += __shfl_down(acc, off, 32);

    __shared__ float red[WAVES_PER_WG];
    if (lane == 0) red[wave] = acc;
    __syncthreads();
    if (threadIdx.x == 0) {
        float s = 0.0f;
        #pragma unroll
        for (int i = 0; i < WAVES_PER_WG; ++i) s += red[i];
        partials[blockIdx.x] = s;
    }
}

__global__ __launch_bounds__(256) void poseloss_finish(
    const float* __restrict__ partials, float* __restrict__ out,
    int n, float inv_batch)
{
    __shared__ float red[256];
    float s = 0.0f;
    for (int i = threadIdx.x; i < n; i += 256) s += partials[i];
    red[threadIdx.x] = s;
    __syncthreads();
    #pragma unroll
    for (int off = 128; off > 0; off >>= 1) {
        if ((int)threadIdx.x < off) red[threadIdx.x] += red[threadIdx.x + off];
        __syncthreads();
    }
    if (threadIdx.x == 0) out[0] = red[0] * inv_batch;
}

extern "C" void kernel_launch(void* const* d_in, const int* in_sizes, int n_in,
                              void* d_out, int out_size, void* d_ws, size_t ws_size,
                              hipStream_t stream)
{
    const float* pred = (const float*)d_in[0];
    const float* targ = (const float*)d_in[1];
    const float* u    = (const float*)d_in[2];
    const float* mean = (const float*)d_in[3];
    float* out = (float*)d_out;
    float* partials = (float*)d_ws;

    const int nrows = in_sizes[0] / DIMS;           // 131072
    const int nblocks = nrows / ROWS_PER_WG;        // 1024 (exact)

    poseloss_main<<<nblocks, 256, 0, stream>>>(pred, targ, u, mean, partials);
    poseloss_finish<<<1, 256, 0, stream>>>(partials, out, nblocks,
                                           1.0f / (float)nrows);
}